// STNeuralPrior_61409442398989
// MI455X (gfx1250) — compile-verified
//
#include <hip/hip_runtime.h>

#define NPTS 8192
#define KNN 16
#define GDIM 256
#define FILT 128
#define GRID_MIN_F (-12.8f)
#define VOXEL_F (0.1f)

#define WSTRIDE 132   // LDS row stride for weights (bank-spread, 16B-aligned rows)
#define HSTRIDE 130   // LDS row stride for activations (bank-spread, 8B-aligned)

typedef __attribute__((ext_vector_type(2))) float v2f;
typedef __attribute__((ext_vector_type(8))) float v8f;
typedef __attribute__((ext_vector_type(4))) int   v4i;

#if defined(__gfx1250__) && __has_builtin(__builtin_amdgcn_global_load_async_to_lds_b128)
#define USE_ASYNC_LDS 1
typedef __attribute__((address_space(1))) v4i* async_gptr_t;  // global int4*
typedef __attribute__((address_space(3))) v4i* async_lptr_t;  // LDS int4*
#else
#define USE_ASYNC_LDS 0
#endif

// Stage one 128x128 fp32 weight matrix into LDS (row stride WSTRIDE).
// Async path: each of 128 threads issues 32 coalesced 16B async global->LDS
// copies (whole block = 64KB), tracked by ASYNCcnt; overlaps with WMMA math.
__device__ __forceinline__ void stage_layer(const float* __restrict__ Wg,
                                            float* dstLds, int t) {
#if USE_ASYNC_LDS
    #pragma unroll
    for (int i = 0; i < 32; ++i) {
        const int c   = t + 128 * i;      // 16-byte chunk id, 0..4095
        const int row = c >> 5;           // 32 chunks per 128-float row
        const int col = (c & 31) << 2;
        __builtin_amdgcn_global_load_async_to_lds_b128(
            (async_gptr_t)(Wg + row * FILT + col),
            (async_lptr_t)(dstLds + row * WSTRIDE + col),
            0, 0);
    }
#else
    for (int e = t; e < FILT * FILT; e += 128) {
        const int k = e >> 7, n = e & 127;
        dstLds[k * WSTRIDE + n] = Wg[e];
    }
#endif
}

__device__ __forceinline__ void wait_async_all() {
#if USE_ASYNC_LDS
#if __has_builtin(__builtin_amdgcn_s_wait_asynccnt)
    __builtin_amdgcn_s_wait_asynccnt(0);
#else
    asm volatile("s_wait_asynccnt 0" ::: "memory");
#endif
#endif
}

// ---------------------------------------------------------------------------
// MLP + rigid flow. 4 waves/block (one per SIMD32 of a WGP), each wave owns
// 16 points. Hidden 128x128 layers run on V_WMMA_F32_16X16X4_F32 (fp32 matrix
// pipe = exact reference precision). Weights are double-buffered in LDS and
// streamed with GLOBAL_LOAD_ASYNC_TO_LDS_B128 one layer ahead of the math.
// ---------------------------------------------------------------------------
__global__ __launch_bounds__(128) void mlp_kernel(
    const float* __restrict__ pc1,
    const float* __restrict__ in_w,  const float* __restrict__ in_b,
    const float* __restrict__ hid_w, const float* __restrict__ hid_b,
    const float* __restrict__ out_w, const float* __restrict__ out_b,
    const float* __restrict__ quat,  const float* __restrict__ trans,
    float* __restrict__ flow_out)
{
    __shared__ __align__(16) float sW[2 * FILT * WSTRIDE];   // double-buffered W[k][n]
    __shared__ __align__(16) float sH[4 * 16 * HSTRIDE];     // per-wave h[16][128]

    const int t    = threadIdx.x;
    const int wave = t >> 5;
    const int lane = t & 31;
    const int m    = lane & 15;     // row within 16-point tile
    const int half = lane >> 4;     // 0: lanes 0-15, 1: lanes 16-31
    const int ptBase = blockIdx.x * 64 + wave * 16;
    float* hW = sH + wave * (16 * HSTRIDE);

    // prefetch layer-0 hidden weights into buffer 0 while we do the input layer
    stage_layer(hid_w, sW, t);

    // ---- input layer: [3 -> 128], too skinny for WMMA, plain VALU ----
    {
        const int pt = ptBase + m;
        const float px = pc1[pt * 3 + 0];
        const float py = pc1[pt * 3 + 1];
        const float pz = pc1[pt * 3 + 2];
        const int n0 = half * 64;
        for (int n = n0; n < n0 + 64; ++n) {
            float v = px * in_w[n] + py * in_w[FILT + n] + pz * in_w[2 * FILT + n] + in_b[n];
            hW[m * HSTRIDE + n] = fmaxf(v, 0.0f);
        }
    }

    // ---- 7 hidden layers: [128 -> 128] via fp32 WMMA, weights double-buffered ----
    int buf = 0;
    for (int l = 0; l < 7; ++l) {
        wait_async_all();      // this wave's issued async copies are complete
        __syncthreads();       // => everyone's copies complete & prev GEMM reads done
        if (l < 6)             // stream next layer while we compute this one
            stage_layer(hid_w + (l + 1) * FILT * FILT, sW + (buf ^ 1) * (FILT * WSTRIDE), t);

        const float* Wb = sW + buf * (FILT * WSTRIDE);

        v8f acc[8];
#pragma unroll
        for (int nt = 0; nt < 8; ++nt)
            acc[nt] = (v8f){0.f, 0.f, 0.f, 0.f, 0.f, 0.f, 0.f, 0.f};

        for (int k0 = 0; k0 < FILT; k0 += 4) {
            const int kb = k0 + 2 * half;
            // A 16x4 fp32 fragment: lanes 0-15 -> K=k0,k0+1 ; lanes 16-31 -> K=k0+2,k0+3
            v2f a = *(const v2f*)(hW + m * HSTRIDE + kb);
#pragma unroll
            for (int nt = 0; nt < 8; ++nt) {
                const int n = nt * 16 + m;       // B 4x16 fragment, symmetric layout
                v2f b;
                b.x = Wb[kb * WSTRIDE + n];
                b.y = Wb[(kb + 1) * WSTRIDE + n];
                acc[nt] = __builtin_amdgcn_wmma_f32_16x16x4_f32(
                    false, a, false, b, (short)0, acc[nt], false, false);
            }
        }

        // epilogue: bias + relu, write back to per-wave h tile
        const float* hb = hid_b + l * FILT;
#pragma unroll
        for (int nt = 0; nt < 8; ++nt) {
            const int n  = nt * 16 + m;
            const float bn = hb[n];
#pragma unroll
            for (int i = 0; i < 8; ++i) {
                const int row = i + 8 * half;  // C/D layout: VGPR i -> M=i / M=i+8
                hW[row * HSTRIDE + n] = fmaxf(acc[nt][i] + bn, 0.0f);
            }
        }
        buf ^= 1;
    }
    __syncthreads();

    // ---- output layer [128 -> 3] + rigid flow, lanes 0-15 per wave ----
    if (half == 0) {
        const int pt = ptBase + m;
        float o0 = out_b[0], o1 = out_b[1], o2 = out_b[2];
        for (int k = 0; k < FILT; ++k) {
            const float h = hW[m * HSTRIDE + k];
            o0 += h * out_w[k * 3 + 0];
            o1 += h * out_w[k * 3 + 1];
            o2 += h * out_w[k * 3 + 2];
        }
        float qw = quat[0], qx = quat[1], qy = quat[2], qz = quat[3];
        const float qn = rsqrtf(qw * qw + qx * qx + qy * qy + qz * qz);
        qw *= qn; qx *= qn; qy *= qn; qz *= qn;
        const float R00 = 1.f - 2.f * (qy * qy + qz * qz), R01 = 2.f * (qx * qy - qw * qz), R02 = 2.f * (qx * qz + qw * qy);
        const float R10 = 2.f * (qx * qy + qw * qz), R11 = 1.f - 2.f * (qx * qx + qz * qz), R12 = 2.f * (qy * qz - qw * qx);
        const float R20 = 2.f * (qx * qz - qw * qy), R21 = 2.f * (qy * qz + qw * qx), R22 = 1.f - 2.f * (qx * qx + qy * qy);
        const float px = pc1[pt * 3 + 0], py = pc1[pt * 3 + 1], pz = pc1[pt * 3 + 2];
        const float r0 = R00 * px + R01 * py + R02 * pz + trans[0] - px;
        const float r1 = R10 * px + R11 * py + R12 * pz + trans[1] - py;
        const float r2 = R20 * px + R21 * py + R22 * pz + trans[2] - pz;
        flow_out[pt * 3 + 0] = o0 + r0;
        flow_out[pt * 3 + 1] = o1 + r1;
        flow_out[pt * 3 + 2] = o2 + r2;
    }
}

// ---------------------------------------------------------------------------
// DT trilinear lookup of warped points; masked sum + count into workspace.
// ---------------------------------------------------------------------------
__global__ __launch_bounds__(256) void dt_kernel(
    const float* __restrict__ pc1, const float* __restrict__ flow,
    const float* __restrict__ grid, float* __restrict__ ws)
{
    const int pt = blockIdx.x * 256 + threadIdx.x;
    float fx[3];
    int   i0[3];
#pragma unroll
    for (int c = 0; c < 3; ++c) {
        const float w = pc1[pt * 3 + c] + flow[pt * 3 + c];
        float ix = (w - GRID_MIN_F) / VOXEL_F;
        ix = fminf(fmaxf(ix, 0.0f), (float)(GDIM - 1) - 1e-4f);
        const int i = (int)floorf(ix);
        i0[c] = i;
        fx[c] = ix - (float)i;
    }
    float s = 0.0f;
#pragma unroll
    for (int dx = 0; dx < 2; ++dx) {
        const float wx = dx ? fx[0] : 1.0f - fx[0];
#pragma unroll
        for (int dy = 0; dy < 2; ++dy) {
            const float wy = dy ? fx[1] : 1.0f - fx[1];
#pragma unroll
            for (int dz = 0; dz < 2; ++dz) {
                const float wz = dz ? fx[2] : 1.0f - fx[2];
                s += wx * wy * wz *
                     grid[((i0[0] + dx) << 16) + ((i0[1] + dy) << 8) + (i0[2] + dz)];
            }
        }
    }
    float contrib = (s < 2.0f) ? s : 0.0f;
    float cnt     = (s < 2.0f) ? 1.0f : 0.0f;
#pragma unroll
    for (int off = 16; off >= 1; off >>= 1) {
        contrib += __shfl_xor(contrib, off, 32);
        cnt     += __shfl_xor(cnt, off, 32);
    }
    if ((threadIdx.x & 31) == 0) {
        atomicAdd(&ws[0], contrib);
        atomicAdd(&ws[2], cnt);
    }
}

// ---------------------------------------------------------------------------
// SC2-KNN: 16 lanes per point, row of A in registers, power iteration via
// width-16 wave shuffles, Rayleigh quotient accumulated atomically.
// ---------------------------------------------------------------------------
__global__ __launch_bounds__(256) void sc_kernel(
    const float* __restrict__ pc1, const float* __restrict__ flow,
    const int* __restrict__ knn, float* __restrict__ ws)
{
    const int point = blockIdx.x * 16 + (threadIdx.x >> 4);
    const int k     = threadIdx.x & 15;
    const int idx   = knn[point * KNN + k];

    const float sx = pc1[idx * 3 + 0];
    const float sy = pc1[idx * 3 + 1];
    const float sz = pc1[idx * 3 + 2];
    const float tx = sx + flow[idx * 3 + 0];
    const float ty = sy + flow[idx * 3 + 1];
    const float tz = sz + flow[idx * 3 + 2];

    const float inv_d2 = 1.0f / (0.03f * 0.03f);
    float Arow[KNN];
#pragma unroll
    for (int j = 0; j < KNN; ++j) {
        const float sxj = __shfl(sx, j, 16), syj = __shfl(sy, j, 16), szj = __shfl(sz, j, 16);
        const float txj = __shfl(tx, j, 16), tyj = __shfl(ty, j, 16), tzj = __shfl(tz, j, 16);
        const float dsx = sx - sxj, dsy = sy - syj, dsz = sz - szj;
        const float dtx = tx - txj, dty = ty - tyj, dtz = tz - tzj;
        const float ds = sqrtf(dsx * dsx + dsy * dsy + dsz * dsz + 1e-8f);
        const float dt = sqrtf(dtx * dtx + dty * dty + dtz * dtz + 1e-8f);
        const float c  = fabsf(ds - dt);
        Arow[j] = fmaxf(0.0f, 1.0f - c * c * inv_d2);
    }

    float v = 0.25f;  // 1/sqrt(16)
    for (int it = 0; it < 10; ++it) {
        float u = 0.0f;
#pragma unroll
        for (int j = 0; j < KNN; ++j)
            u += Arow[j] * __shfl(v, j, 16);
        float s = u * u;
#pragma unroll
        for (int off = 8; off >= 1; off >>= 1)
            s += __shfl_xor(s, off, 16);
        v = u / (sqrtf(s) + 1e-6f);
    }
    float u = 0.0f;
#pragma unroll
    for (int j = 0; j < KNN; ++j)
        u += Arow[j] * __shfl(v, j, 16);
    float e = v * u;
#pragma unroll
    for (int off = 8; off >= 1; off >>= 1)
        e += __shfl_xor(e, off, 16);

    if (k == 0) atomicAdd(&ws[1], e);
}

__global__ void init_ws(float* ws) {
    ws[0] = 0.0f;  // dt masked sum
    ws[1] = 0.0f;  // eig sum
    ws[2] = 0.0f;  // mask count
}

__global__ void finalize_kernel(const float* __restrict__ ws, float* __restrict__ loss_out) {
    const float dt_loss = ws[0] / fmaxf(ws[2], 1.0f);
    const float sc_loss = -(ws[1] / (float)NPTS) / (float)KNN;
    loss_out[0] = dt_loss + sc_loss;
}

extern "C" void kernel_launch(void* const* d_in, const int* in_sizes, int n_in,
                              void* d_out, int out_size, void* d_ws, size_t ws_size,
                              hipStream_t stream) {
    (void)in_sizes; (void)n_in; (void)out_size; (void)ws_size;
    const float* pc1     = (const float*)d_in[0];
    const float* dt_grid = (const float*)d_in[1];
    const float* in_w    = (const float*)d_in[2];
    const float* in_b    = (const float*)d_in[3];
    const float* hid_w   = (const float*)d_in[4];
    const float* hid_b   = (const float*)d_in[5];
    const float* out_w   = (const float*)d_in[6];
    const float* out_b   = (const float*)d_in[7];
    const float* quat    = (const float*)d_in[8];
    const float* trans   = (const float*)d_in[9];
    const int*   knn     = (const int*)d_in[10];
    float* flow = (float*)d_out;            // [0 .. 24575] pred_flow, [24576] loss
    float* ws   = (float*)d_ws;

    init_ws<<<1, 1, 0, stream>>>(ws);
    mlp_kernel<<<NPTS / 64, 128, 0, stream>>>(pc1, in_w, in_b, hid_w, hid_b,
                                              out_w, out_b, quat, trans, flow);
    dt_kernel<<<NPTS / 256, 256, 0, stream>>>(pc1, flow, dt_grid, ws);
    sc_kernel<<<NPTS / 16, 256, 0, stream>>>(pc1, flow, knn, ws);
    finalize_kernel<<<1, 1, 0, stream>>>(ws, flow + NPTS * 3);
}